// Set2Set_81003083203037
// MI455X (gfx1250) — compile-verified
//
#include <hip/hip_runtime.h>
#include <hip/hip_bf16.h>

// ---------------------------------------------------------------------------
// Set2Set (N=400000 nodes, F=256, B=1024 graphs, 3 steps)
// Strategy:
//   * LSTM GEMM gates[1024,1024] = q_star @ W_ih^T + h @ W_hh^T  via
//     V_WMMA_F32_16X16X4_F32 (one wave per 16x16 tile, K-step 4).
//   * Attention: ONE pass over x per step using online softmax
//     (3 x 409.6MB = 1.23GB HBM traffic -> ~53us floor at 23.3TB/s).
//     One 256-thread (8-wave) workgroup per graph; batch is sorted so each
//     graph's nodes are contiguous; per-node dot via wave32 shfl reduction.
// ---------------------------------------------------------------------------

#define FDIM   256
#define BGRAPH 1024
#define K_IH   512      // 2F
#define K_HH   256      // F
#define GCOLS  1024     // 4F
#define QSTARW 512      // 2F

typedef float v2f __attribute__((ext_vector_type(2)));
typedef float v8f __attribute__((ext_vector_type(8)));

// ---------------- init: zero state, sentinel starts ----------------
__global__ void init_state_k(float* h, float* c, float* q_star, int* starts, int n_nodes) {
    int t = blockIdx.x * blockDim.x + threadIdx.x;
    if (t < BGRAPH * FDIM) { h[t] = 0.f; c[t] = 0.f; }
    if (t < BGRAPH * QSTARW) q_star[t] = 0.f;
    if (t <= BGRAPH) starts[t] = n_nodes;   // sentinel; atomicMin below
}

// ---------------- segment starts (batch is sorted) ----------------
__global__ void seg_starts_k(const int* __restrict__ batch, int* starts, int n_nodes) {
    int i = blockIdx.x * blockDim.x + threadIdx.x;
    if (i >= n_nodes) return;
    atomicMin(&starts[batch[i]], i);
}

__global__ void seg_fix_k(int* starts) {
    if (threadIdx.x == 0 && blockIdx.x == 0) {
        // empty graphs inherit the next start so their range is empty
        for (int g = BGRAPH - 1; g >= 0; --g)
            if (starts[g] > starts[g + 1]) starts[g] = starts[g + 1];
    }
}

// ---------------- LSTM GEMM via f32 WMMA ----------------
// gates[m][n] = sum_k q_star[m][k] * W_ih[n][k]  +  sum_k h[m][k] * W_hh[n][k]
// (B-matrix of the WMMA is W^T, so its fragment is a contiguous float2 of W's row n)
__global__ __launch_bounds__(256) void lstm_gemm_k(
        const float* __restrict__ q_star,  // [1024,512]
        const float* __restrict__ h,       // [1024,256]
        const float* __restrict__ W_ih,    // [1024,512]
        const float* __restrict__ W_hh,    // [1024,256]
        float* __restrict__ gates)         // [1024,1024]
{
    const int wave = threadIdx.x >> 5;
    const int lane = threadIdx.x & 31;
    const int tile = blockIdx.x * 8 + wave;        // 4096 tiles total
    const int mt = tile >> 6;                      // 64 tiles along M
    const int nt = tile & 63;                      // 64 tiles along N
    const int m0 = mt * 16, n0 = nt * 16;
    const int ln    = lane & 15;
    const int khalf = (lane >> 4) * 2;             // lanes 16-31 hold K+2,K+3

    v8f acc = {};

    // --- K = 512 over W_ih ---
    const float* Arow = q_star + (size_t)(m0 + ln) * K_IH;
    const float* Brow = W_ih   + (size_t)(n0 + ln) * K_IH;
    for (int k0 = 0; k0 < K_IH; k0 += 4) {
        v2f a = *(const v2f*)(Arow + k0 + khalf);
        v2f b = *(const v2f*)(Brow + k0 + khalf);
        acc = __builtin_amdgcn_wmma_f32_16x16x4_f32(
                  false, a, false, b, (short)0, acc, false, false);
    }
    // --- K = 256 over W_hh ---
    const float* Arow2 = h    + (size_t)(m0 + ln) * K_HH;
    const float* Brow2 = W_hh + (size_t)(n0 + ln) * K_HH;
    for (int k0 = 0; k0 < K_HH; k0 += 4) {
        v2f a = *(const v2f*)(Arow2 + k0 + khalf);
        v2f b = *(const v2f*)(Brow2 + k0 + khalf);
        acc = __builtin_amdgcn_wmma_f32_16x16x4_f32(
                  false, a, false, b, (short)0, acc, false, false);
    }

    // C/D layout: VGPR r -> row m0 + r + 8*(lane>=16), col n0 + lane%16
    const int rbase = m0 + (lane >> 4) * 8;
    const int col   = n0 + ln;
    #pragma unroll
    for (int r = 0; r < 8; ++r)
        gates[(size_t)(rbase + r) * GCOLS + col] = acc[r];
}

// ---------------- gate nonlinearities ----------------
__device__ __forceinline__ float sigf(float v) { return 1.f / (1.f + __expf(-v)); }

__global__ void lstm_gate_k(const float* __restrict__ gates,
                            const float* __restrict__ b_ih,
                            const float* __restrict__ b_hh,
                            float* __restrict__ c,
                            float* __restrict__ h,
                            float* __restrict__ q_star)
{
    int t = blockIdx.x * blockDim.x + threadIdx.x;
    if (t >= BGRAPH * FDIM) return;
    int g = t >> 8, f = t & 255;
    const float* G = gates + (size_t)g * GCOLS;
    float iv = G[f]           + b_ih[f]           + b_hh[f];
    float fv = G[256 + f]     + b_ih[256 + f]     + b_hh[256 + f];
    float gv = G[512 + f]     + b_ih[512 + f]     + b_hh[512 + f];
    float ov = G[768 + f]     + b_ih[768 + f]     + b_hh[768 + f];
    float cn = sigf(fv) * c[t] + sigf(iv) * tanhf(gv);
    float qv = sigf(ov) * tanhf(cn);
    c[t] = cn;
    h[t] = qv;                               // h := q
    q_star[(size_t)g * QSTARW + f] = qv;     // q_star[:, :F] = q
}

// ---------------- online-softmax segment attention ----------------
// One 8-wave block per graph. Each lane owns 8 contiguous channels.
__global__ __launch_bounds__(256) void attn_k(const float* __restrict__ x,
                                              const int* __restrict__ starts,
                                              const float* __restrict__ h,  // q [B,F]
                                              float* __restrict__ q_star)
{
    const int g    = blockIdx.x;
    const int tid  = threadIdx.x;
    const int wave = tid >> 5;
    const int lane = tid & 31;

    __shared__ float q_sh[FDIM];
    __shared__ float m_sh[8], s_sh[8];
    __shared__ float r_sh[8 * FDIM];

    q_sh[tid] = h[(size_t)g * FDIM + tid];
    __syncthreads();

    float qreg[8];
    #pragma unroll
    for (int j = 0; j < 8; ++j) qreg[j] = q_sh[lane * 8 + j];

    const int s0 = starts[g], s1 = starts[g + 1];
    float m = -__builtin_inff();
    float s = 0.f;
    float racc[8];
    #pragma unroll
    for (int j = 0; j < 8; ++j) racc[j] = 0.f;

    for (int n = s0 + wave; n < s1; n += 8) {
        const float* xr = x + (size_t)n * FDIM + lane * 8;
        if (n + 8 < s1) __builtin_prefetch(xr + 8 * FDIM, 0, 0); // global_prefetch_b8
        float4 xa = *(const float4*)xr;
        float4 xb = *(const float4*)(xr + 4);
        float p = xa.x * qreg[0] + xa.y * qreg[1] + xa.z * qreg[2] + xa.w * qreg[3]
                + xb.x * qreg[4] + xb.y * qreg[5] + xb.z * qreg[6] + xb.w * qreg[7];
        #pragma unroll
        for (int off = 16; off; off >>= 1) p += __shfl_xor(p, off, 32);
        float e = p;   // full dot, identical in all lanes

        if (e <= m) {
            float w = __expf(e - m);
            s += w;
            racc[0] += w * xa.x; racc[1] += w * xa.y; racc[2] += w * xa.z; racc[3] += w * xa.w;
            racc[4] += w * xb.x; racc[5] += w * xb.y; racc[6] += w * xb.z; racc[7] += w * xb.w;
        } else {
            float sc = __expf(m - e);   // exp(-inf)=0 handles first node
            s = s * sc + 1.f;
            racc[0] = racc[0] * sc + xa.x; racc[1] = racc[1] * sc + xa.y;
            racc[2] = racc[2] * sc + xa.z; racc[3] = racc[3] * sc + xa.w;
            racc[4] = racc[4] * sc + xb.x; racc[5] = racc[5] * sc + xb.y;
            racc[6] = racc[6] * sc + xb.z; racc[7] = racc[7] * sc + xb.w;
            m = e;
        }
    }

    if (lane == 0) { m_sh[wave] = m; s_sh[wave] = s; }
    #pragma unroll
    for (int j = 0; j < 8; ++j) r_sh[wave * FDIM + lane * 8 + j] = racc[j];
    __syncthreads();

    // cross-wave combine: thread tid owns channel tid
    float M = -__builtin_inff();
    #pragma unroll
    for (int w = 0; w < 8; ++w) M = fmaxf(M, m_sh[w]);
    float S = 0.f, R = 0.f;
    #pragma unroll
    for (int w = 0; w < 8; ++w) {
        float sc = (m_sh[w] == -__builtin_inff()) ? 0.f : __expf(m_sh[w] - M);
        S += s_sh[w] * sc;
        R += r_sh[w * FDIM + tid] * sc;
    }
    float r = (S > 0.f) ? (R / S) : 0.f;
    q_star[(size_t)g * QSTARW + FDIM + tid] = r;   // q_star[:, F:2F] = r
}

// ---------------- final copy ----------------
__global__ void copy_out_k(const float* __restrict__ q_star, float* __restrict__ out) {
    int t = blockIdx.x * blockDim.x + threadIdx.x;
    if (t < BGRAPH * QSTARW) out[t] = q_star[t];
}

// ---------------------------------------------------------------------------
extern "C" void kernel_launch(void* const* d_in, const int* in_sizes, int n_in,
                              void* d_out, int out_size, void* d_ws, size_t ws_size,
                              hipStream_t stream) {
    const float* x     = (const float*)d_in[0];
    const int*   batch = (const int*)  d_in[1];
    const float* W_ih  = (const float*)d_in[2];
    const float* W_hh  = (const float*)d_in[3];
    const float* b_ih  = (const float*)d_in[4];
    const float* b_hh  = (const float*)d_in[5];
    const int n_nodes  = in_sizes[1];

    // workspace layout (floats): h | c | q_star | gates | starts(int)
    float* ws     = (float*)d_ws;
    float* h      = ws;                                  // 262144
    float* c      = h + BGRAPH * FDIM;                   // 262144
    float* q_star = c + BGRAPH * FDIM;                   // 524288
    float* gates  = q_star + BGRAPH * QSTARW;            // 1048576
    int*   starts = (int*)(gates + (size_t)BGRAPH * GCOLS); // 1025 ints (~8.4MB total)

    const int T = 256;
    init_state_k<<<(BGRAPH * QSTARW + T - 1) / T, T, 0, stream>>>(h, c, q_star, starts, n_nodes);
    seg_starts_k<<<(n_nodes + T - 1) / T, T, 0, stream>>>(batch, starts, n_nodes);
    seg_fix_k<<<1, 32, 0, stream>>>(starts);

    for (int step = 0; step < 3; ++step) {
        lstm_gemm_k<<<512, 256, 0, stream>>>(q_star, h, W_ih, W_hh, gates);
        lstm_gate_k<<<(BGRAPH * FDIM + T - 1) / T, T, 0, stream>>>(gates, b_ih, b_hh, c, h, q_star);
        attn_k<<<BGRAPH, 256, 0, stream>>>(x, starts, h, q_star);
    }
    copy_out_k<<<(BGRAPH * QSTARW + T - 1) / T, T, 0, stream>>>(q_star, (float*)d_out);
}